// MiniBatchAUC_68685116998328
// MI455X (gfx1250) — compile-verified
//
#include <hip/hip_runtime.h>
#include <hip/hip_bf16.h>

typedef __attribute__((ext_vector_type(2))) float v2f;
typedef __attribute__((ext_vector_type(8))) float v8f;

#define N_ELEMS 16384
#define BLOCK   1024

// Full 32-lane sum via V_WMMA_F32_16X16X4_F32.
// A-matrix 16x4 layout (ISA 7.12.2): lanes 0-15 hold (K=0,K=1) of row M=lane,
// lanes 16-31 hold (K=2,K=3) of row M=lane-16. With a1=0 and B=all-ones:
//   D[m][n] = v(lane m) + v(lane m+16)   (all-ones B is layout-invariant)
// D layout: VGPR r in lane L holds row M = r + 8*(L>=16). Summing d[0..7]
// gives lanes 0-15 -> sum of lanes {0..7,16..23}, lanes 16-31 -> the rest;
// one shfl_xor(16) completes the reduction (result in every lane).
__device__ __forceinline__ float wave_allsum_wmma(float v) {
    v2f a; a[0] = v;    a[1] = 0.0f;
    v2f b; b[0] = 1.0f; b[1] = 1.0f;
    v8f c = {};
    v8f d = __builtin_amdgcn_wmma_f32_16x16x4_f32(
        /*neg_a=*/false, a, /*neg_b=*/false, b,
        /*c_mod=*/(short)0, c, /*reuse_a=*/false, /*reuse_b=*/false);
    float t = ((d[0] + d[1]) + (d[2] + d[3])) + ((d[4] + d[5]) + (d[6] + d[7]));
    t += __shfl_xor(t, 16, 32);
    return t;
}

__global__ __launch_bounds__(BLOCK)
void MiniBatchAUC_kernel(const float* __restrict__ logits,
                         const long long* __restrict__ targets,
                         float* __restrict__ out) {
    __shared__ float part[5][32];
    const int tid = threadIdx.x;

    // Per-lane partial statistics.
    float S = 0.0f, Q = 0.0f, Sp = 0.0f, Qp = 0.0f, np = 0.0f;

    const float4* lg4 = (const float4*)logits;   // 4096 float4
    const int4*   tg4 = (const int4*)targets;    // 8192 int4 (2 int64 each)

    #pragma unroll
    for (int k = tid; k < N_ELEMS / 4; k += BLOCK) {   // 4 iterations
        float4 x  = lg4[k];
        int4   t0 = tg4[2 * k + 0];
        int4   t1 = tg4[2 * k + 1];
        float xs[4] = { x.x, x.y, x.z, x.w };
        int   ps[4] = { (t0.x | t0.y), (t0.z | t0.w),
                        (t1.x | t1.y), (t1.z | t1.w) };
        #pragma unroll
        for (int j = 0; j < 4; ++j) {
            float s = __fdividef(1.0f, 1.0f + __expf(-xs[j]));  // sigmoid
            float q = s * s;
            float p = (ps[j] != 0) ? 1.0f : 0.0f;
            S += s;
            Q += q;
            Sp = fmaf(p, s, Sp);
            Qp = fmaf(p, q, Qp);
            np += p;
        }
    }

    // Stage 1: 32-lane reduction per wave on the matrix pipe.
    S  = wave_allsum_wmma(S);
    Q  = wave_allsum_wmma(Q);
    Sp = wave_allsum_wmma(Sp);
    Qp = wave_allsum_wmma(Qp);
    np = wave_allsum_wmma(np);

    const int lane = tid & 31;
    const int wid  = tid >> 5;
    if (lane == 0) {
        part[0][wid] = S;  part[1][wid] = Q;  part[2][wid] = Sp;
        part[3][wid] = Qp; part[4][wid] = np;
    }
    __syncthreads();

    // Stage 2: wave 0 reduces the 32 wave partials (again via WMMA).
    if (tid < 32) {
        float s2  = wave_allsum_wmma(part[0][tid]);
        float q2  = wave_allsum_wmma(part[1][tid]);
        float sp2 = wave_allsum_wmma(part[2][tid]);
        float qp2 = wave_allsum_wmma(part[3][tid]);
        float np2 = wave_allsum_wmma(part[4][tid]);
        if (tid == 0) {
            float n_pos = np2;
            float n_neg = (float)N_ELEMS - n_pos;
            float Sn = s2 - sp2;
            float Qn = q2 - qp2;
            // sum over pos i, neg j of (1 - s_i + s_j)^2
            float loss = n_pos * n_neg
                       + n_neg * qp2 + n_pos * Qn
                       - 2.0f * n_neg * sp2 + 2.0f * n_pos * Sn
                       - 2.0f * sp2 * Sn;
            out[0] = loss / (n_pos * n_neg);
        }
    }
}

extern "C" void kernel_launch(void* const* d_in, const int* in_sizes, int n_in,
                              void* d_out, int out_size, void* d_ws, size_t ws_size,
                              hipStream_t stream) {
    const float*     logits  = (const float*)d_in[0];
    const long long* targets = (const long long*)d_in[1];
    float*           out     = (float*)d_out;
    hipLaunchKernelGGL(MiniBatchAUC_kernel, dim3(1), dim3(BLOCK), 0, stream,
                       logits, targets, out);
}